// MaxPoolingWord_1872605741239
// MI455X (gfx1250) — compile-verified
//
#include <hip/hip_runtime.h>
#include <stdint.h>

// Problem constants (from the reference): B=16, S=4096, D=1024, N=1024.
// bounds are sorted and spans = (bounds[:-1], bounds[1:]), so the spans are a
// contiguous partition of [0,S): the searchsorted/segment_max construction
// reduces exactly to a per-span max over rows [start, end), zeros if empty.
constexpr int Bc = 16;
constexpr int Sc = 4096;
constexpr int Dc = 1024;        // floats per row; 256 threads * float4
constexpr int Nc = 1024;
constexpr int CHUNK = 8;        // async rows in flight per workgroup
constexpr int ROW_BYTES = Dc * 4;   // 4096 B: global row stride == LDS buffer stride

// ---------------------------------------------------------------------------
// Kernel 1: one workgroup (256 threads, 8 wave32) per (b, n) span.
// Issues up to 8 independent global->LDS async row copies (ASYNCcnt-tracked),
// waits once, then reduces the chunk from LDS with v_max_num_f32.
// Since INST_OFFSET is added to BOTH the global and LDS addresses and both
// strides are 4096 B, all 8 loads share one address pair + immediate offsets.
// ---------------------------------------------------------------------------
__global__ __launch_bounds__(256) void seg_max_kernel(
    const float* __restrict__ ctx,
    const int*   __restrict__ spans,
    float*       __restrict__ out)
{
    __shared__ __align__(16) float buf[CHUNK][Dc];   // 32 KB

    const int wg  = blockIdx.x;          // 0 .. B*N-1
    const int b   = wg >> 10;            // wg / Nc   (Nc == 1024)
    const int n   = wg & (Nc - 1);       // wg % Nc
    const int tid = threadIdx.x;         // 0..255 -> one 16B chunk of the row

    const int2 se   = ((const int2*)spans)[b * Nc + n];
    const int start = se.x;
    const int end   = se.y;

    float4* orow = (float4*)(out + ((size_t)b * Sc + n) * Dc) + tid;

    if (end <= start) {                  // empty span -> zeros (matches ref)
        *orow = make_float4(0.f, 0.f, 0.f, 0.f);
        return;
    }

    // Per-thread base of its 16B slice in row `start`, and its LDS slot.
    const float*   gbase   = ctx + ((size_t)b * Sc + start) * Dc + (tid << 2);
    const unsigned ldsbase = (unsigned)(size_t)(&buf[0][tid << 2]);

    float4 acc = make_float4(-__builtin_inff(), -__builtin_inff(),
                             -__builtin_inff(), -__builtin_inff());

    const int len = end - start;
    for (int done = 0; done < len; done += CHUNK) {
        const int c = (len - done < CHUNK) ? (len - done) : CHUNK;
        const float* gchunk = gbase + (size_t)done * Dc;

        // WAR: LDS reads of the previous chunk must be complete before the
        // async engine overwrites the buffers.
        asm volatile("s_wait_dscnt 0" ::: "memory");

        // Issue c independent async row copies; row j goes to buf[j] via the
        // shared immediate offset (applies to both LDS and global address).
#pragma unroll
        for (int j = 0; j < CHUNK; ++j) {
            if (j < c) {
                asm volatile("global_load_async_to_lds_b128 %0, %1, off offset:%2"
                             :: "v"(ldsbase), "v"(gchunk), "i"(j * ROW_BYTES)
                             : "memory");
            }
        }

        // All c rows landed in LDS.
        asm volatile("s_wait_asynccnt 0" ::: "memory");

#pragma unroll
        for (int j = 0; j < CHUNK; ++j) {
            if (j < c) {
                const float4 v = *(const float4*)&buf[j][tid << 2];  // ds_load_b128
                acc.x = fmaxf(acc.x, v.x);
                acc.y = fmaxf(acc.y, v.y);
                acc.z = fmaxf(acc.z, v.z);
                acc.w = fmaxf(acc.w, v.w);
            }
        }
    }

    *orow = acc;
}

// ---------------------------------------------------------------------------
// Kernel 2: zero-fill out[:, N:S, :]  (B * (S-N) * D floats).
// ---------------------------------------------------------------------------
__global__ __launch_bounds__(256) void zero_tail_kernel(float* __restrict__ out)
{
    const size_t per_batch4 = (size_t)(Sc - Nc) * Dc / 4;   // float4 per batch
    size_t i = (size_t)blockIdx.x * blockDim.x + threadIdx.x;
    size_t b = i / per_batch4;
    size_t r = i - b * per_batch4;
    if (b < (size_t)Bc) {
        float4* p = (float4*)(out + ((size_t)b * Sc + Nc) * Dc) + r;
        *p = make_float4(0.f, 0.f, 0.f, 0.f);
    }
}

// ---------------------------------------------------------------------------
// Launch
// ---------------------------------------------------------------------------
extern "C" void kernel_launch(void* const* d_in, const int* in_sizes, int n_in,
                              void* d_out, int out_size, void* d_ws, size_t ws_size,
                              hipStream_t stream)
{
    (void)in_sizes; (void)n_in; (void)out_size; (void)d_ws; (void)ws_size;

    const float* ctx   = (const float*)d_in[0];   // (B, S, D) f32
    const int*   spans = (const int*)d_in[1];     // (B, N, 2) i32
    float*       out   = (float*)d_out;           // (B, S, D) f32

    const size_t tail4   = (size_t)Bc * (Sc - Nc) * Dc / 4;
    const int    zblocks = (int)((tail4 + 255) / 256);
    zero_tail_kernel<<<zblocks, 256, 0, stream>>>(out);

    seg_max_kernel<<<Bc * Nc, 256, 0, stream>>>(ctx, spans, out);
}